// Attention_40716289966507
// MI455X (gfx1250) — compile-verified
//
#include <hip/hip_runtime.h>
#include <hip/hip_bf16.h>

typedef float v2f __attribute__((ext_vector_type(2)));
typedef float v8f __attribute__((ext_vector_type(8)));
typedef unsigned int v4u __attribute__((ext_vector_type(4)));
typedef int v4i __attribute__((ext_vector_type(4)));
typedef int v8i __attribute__((ext_vector_type(8)));

// D = A(16x4, f32) * B(4x16, f32) + C(16x16, f32)  -> V_WMMA_F32_16X16X4_F32
#define WMMA_F32(a, b, c) \
  __builtin_amdgcn_wmma_f32_16x16x4_f32(false, (a), false, (b), (short)0, (c), false, false)

constexpr int   kB  = 4;
constexpr int   kC  = 256;
constexpr int   kNH = 8;
constexpr int   kHD = 32;          // head dim
constexpr int   kN  = 48 * 48;     // 2304 tokens
constexpr int   kJB = 32;          // key-block width (2 x 16), double-buffered
constexpr float kScale = 0.17677669529663687f; // 32^-0.5

__device__ __forceinline__ uint32_t lds_addr_of(const void* p) {
  // addrspace(3) offsets live in the low 32 bits of the generic pointer
  return (uint32_t)(uintptr_t)p;
}

// ---------------------------------------------------------------------------
// TDM: async DMA of a 2D row-major tile (global, stride_elems f32 per row)
// into LDS. pad32=true inserts 2 DWORDs of LDS padding every 32 DWORDs
// (row stride 34 for 32-wide tiles -> conflict-free column reads).
// Tracked by TENSORcnt; caller pairs with s_wait_tensorcnt + barrier.
// ---------------------------------------------------------------------------
__device__ __forceinline__ void tdm_load_2d(uint32_t lds_off, const float* gptr,
                                            uint32_t tile_w, uint32_t tile_h,
                                            uint32_t stride_elems, bool pad32)
{
  const uint64_t ga = (uint64_t)(uintptr_t)gptr;
  v4u g0;
  g0[0] = 1u;                                                 // count=1 (valid)
  g0[1] = lds_off;                                            // LDS byte address
  g0[2] = (uint32_t)ga;                                       // global_addr[31:0]
  g0[3] = (uint32_t)((ga >> 32) & 0x1FFFFFFu) | (2u << 30);   // ga[56:32] | type=2

  uint32_t w0 = 2u << 16;                                     // data_size = 4B
  if (pad32) w0 |= (1u << 20) | (4u << 22) | (1u << 25);      // pad_en, 32dw, 2dw

  v8i g1;
  g1[0] = (int)w0;
  g1[1] = (int)((stride_elems & 0xFFFFu) << 16);              // tensor_dim0 lo16
  g1[2] = (int)(((stride_elems >> 16) & 0xFFFFu) | (tile_h << 16)); // dim0 hi | dim1 lo
  g1[3] = (int)(tile_w << 16);                                // dim1 hi=0 | tile_dim0
  g1[4] = (int)(tile_h & 0xFFFFu);                            // tile_dim1 | tile_dim2=0
  g1[5] = (int)stride_elems;                                  // dim0_stride lo32
  g1[6] = 0;                                                  // stride hi | dim1_stride
  g1[7] = 0;
  v4i z4 = {0, 0, 0, 0};
  v8i z8 = {0, 0, 0, 0, 0, 0, 0, 0};
  // 6-arg toolchain variant: (g0, g1, g2, g3, g4, cpol)
  __builtin_amdgcn_tensor_load_to_lds(g0, g1, z4, z4, z8, 0);
}

// ---------------------------------------------------------------------------
// Y[b][m][n] = sum_c W[m][c] * X[b][c][n] (+ bias[m])
// block tile 64(M) x 128(N), 8 waves (each 16x64 = 4 WMMA tiles),
// K-chunks of 32, TDM double-buffered.  grid = (kN/128, M/64, kB)
// ---------------------------------------------------------------------------
__global__ __launch_bounds__(256)
void gemm_wmma(const float* __restrict__ W, const float* __restrict__ X,
               const float* __restrict__ bias, float* __restrict__ Y, int M)
{
  __shared__ float lds_w[2][64 * 34];     // TDM-padded stride 34
  __shared__ float lds_x[2][32 * 128];

  const int n0 = blockIdx.x * 128;
  const int m0 = blockIdx.y * 64;
  const int b  = blockIdx.z;
  const int tid = threadIdx.x;
  const int wave = tid >> 5, lane = tid & 31;
  const int half = lane >> 4, l16 = lane & 15;
  const int mstrip = (wave & 3) * 16;
  const int nstrip = (wave >> 2) * 64;

  const float* Xb = X + (size_t)b * kC * kN;
  const float* Wb = W + (size_t)m0 * kC;

  if (wave == 0) {                       // prefetch chunk 0
    tdm_load_2d(lds_addr_of(lds_w[0]), Wb,      32, 64, kC, true);
    tdm_load_2d(lds_addr_of(lds_x[0]), Xb + n0, 128, 32, kN, false);
  }

  v8f acc[4];
  #pragma unroll
  for (int t = 0; t < 4; ++t) {
    v8f z = {0.f,0.f,0.f,0.f,0.f,0.f,0.f,0.f};
    acc[t] = z;
  }

  constexpr int kSteps = kC / 32;        // 8
  for (int step = 0; step < kSteps; ++step) {
    const int buf = step & 1;
    if (wave == 0) __builtin_amdgcn_s_wait_tensorcnt(0);   // chunk `step` landed
    __syncthreads();                     // publish; everyone done with buf^1
    if (wave == 0 && step + 1 < kSteps) {                  // DMA chunk step+1
      const int c1 = (step + 1) * 32;
      tdm_load_2d(lds_addr_of(lds_w[buf ^ 1]), Wb + c1,                   32, 64, kC, true);
      tdm_load_2d(lds_addr_of(lds_x[buf ^ 1]), Xb + (size_t)c1 * kN + n0, 128, 32, kN, false);
    }
    const float* lw = lds_w[buf];
    const float* lx = lds_x[buf];
    #pragma unroll
    for (int kk = 0; kk < 32; kk += 4) {
      v2f a;                                               // A: lane=M, K=kk+2*half+{0,1}
      a.x = lw[(mstrip + l16) * 34 + kk + half * 2 + 0];
      a.y = lw[(mstrip + l16) * 34 + kk + half * 2 + 1];
      #pragma unroll
      for (int t = 0; t < 4; ++t) {
        v2f bb;                                            // B: lane=N
        bb.x = lx[(kk + half * 2 + 0) * 128 + nstrip + t * 16 + l16];
        bb.y = lx[(kk + half * 2 + 1) * 128 + nstrip + t * 16 + l16];
        acc[t] = WMMA_F32(a, bb, acc[t]);
      }
    }
  }

  // D layout: lane -> N = l16 (+16t), VGPR v -> M = v + 8*half
  #pragma unroll
  for (int t = 0; t < 4; ++t) {
    const int n = n0 + nstrip + t * 16 + l16;
    #pragma unroll
    for (int v = 0; v < 8; ++v) {
      const int m = m0 + mstrip + v + 8 * half;
      float r = acc[t][v];
      if (bias) r += bias[m];
      Y[((size_t)b * M + m) * kN + n] = r;
    }
  }
}

// ---------------------------------------------------------------------------
// Fused flash attention per (i-tile, head, batch); qkv layout [b][3*256][N].
// 128 query rows per block; each wave owns one 16-row strip.
// K/V blocks of 32 keys, TDM double-buffered; P stays in LDS.
// ---------------------------------------------------------------------------
__global__ __launch_bounds__(256)
void attn_wmma(const float* __restrict__ qkv, float* __restrict__ ctx)
{
  __shared__ float lds_q[kHD * 128];      // q[d][i], stride 128       16.0 KB
  __shared__ float lds_k[2][kHD * 34];    // k[d][j], TDM stride 34     8.5 KB
  __shared__ float lds_v[2][kHD * 34];    // v[d][j], TDM stride 34     8.5 KB
  __shared__ float lds_p[128 * 33];       // P strips / O staging      16.5 KB

  const int i0  = blockIdx.x * 128;
  const int h   = blockIdx.y;
  const int b   = blockIdx.z;
  const int tid = threadIdx.x;
  const int wave = tid >> 5, lane = tid & 31;
  const int half = lane >> 4, l16 = lane & 15;
  const int istrip = wave * 16;

  const float* qb = qkv + ((size_t)(b * 3 + 0) * kC + h * kHD) * kN;
  const float* kb = qkv + ((size_t)(b * 3 + 1) * kC + h * kHD) * kN;
  const float* vb = qkv + ((size_t)(b * 3 + 2) * kC + h * kHD) * kN;

  if (wave == 0) {                        // prefetch Q + first K/V block
    tdm_load_2d(lds_addr_of(lds_q),     qb + i0, 128, kHD, kN, false);
    tdm_load_2d(lds_addr_of(lds_k[0]),  kb,      kJB, kHD, kN, true);
    tdm_load_2d(lds_addr_of(lds_v[0]),  vb,      kJB, kHD, kN, true);
  }

  v8f acc0 = {0.f,0.f,0.f,0.f,0.f,0.f,0.f,0.f};   // O cols d=0..15
  v8f acc1 = {0.f,0.f,0.f,0.f,0.f,0.f,0.f,0.f};   // O cols d=16..31
  float mrow[8], lrow[8];
  #pragma unroll
  for (int v = 0; v < 8; ++v) { mrow[v] = -__builtin_inff(); lrow[v] = 0.f; }

  constexpr int kJSteps = kN / kJB;       // 72
  for (int jb = 0; jb < kJSteps; ++jb) {
    const int buf = jb & 1;
    if (wave == 0) __builtin_amdgcn_s_wait_tensorcnt(0);   // block jb landed
    __syncthreads();                      // publish; all done with buf^1
    if (wave == 0 && jb + 1 < kJSteps) {  // DMA block jb+1 under the math
      const int j1 = (jb + 1) * kJB;
      tdm_load_2d(lds_addr_of(lds_k[buf ^ 1]), kb + j1, kJB, kHD, kN, true);
      tdm_load_2d(lds_addr_of(lds_v[buf ^ 1]), vb + j1, kJB, kHD, kN, true);
    }
    const float* lk = lds_k[buf];
    const float* lv = lds_v[buf];

    // ---- S = (Q^T)(K): M=i, K-dim=d(32), N=j(32): 2 j-tiles x 8 k-steps ----
    v8f st[2];
    #pragma unroll
    for (int jt = 0; jt < 2; ++jt) {
      v8f z = {0.f,0.f,0.f,0.f,0.f,0.f,0.f,0.f};
      st[jt] = z;
    }
    #pragma unroll
    for (int kk = 0; kk < kHD; kk += 4) {
      v2f a;
      a.x = lds_q[(kk + half * 2 + 0) * 128 + istrip + l16];
      a.y = lds_q[(kk + half * 2 + 1) * 128 + istrip + l16];
      #pragma unroll
      for (int jt = 0; jt < 2; ++jt) {
        v2f bb;
        bb.x = lk[(kk + half * 2 + 0) * 34 + jt * 16 + l16];
        bb.y = lk[(kk + half * 2 + 1) * 34 + jt * 16 + l16];
        st[jt] = WMMA_F32(a, bb, st[jt]);
      }
    }
    st[0] = st[0] * kScale;
    st[1] = st[1] * kScale;

    // ---- online softmax: row stats live in the 16 lanes holding that row ----
    float alpha[8];
    #pragma unroll
    for (int v = 0; v < 8; ++v) {
      float mb = fmaxf(st[0][v], st[1][v]);
      #pragma unroll
      for (int off = 8; off >= 1; off >>= 1)
        mb = fmaxf(mb, __shfl_xor(mb, off, 16));
      const float mn = fmaxf(mrow[v], mb);
      const float a  = __expf(mrow[v] - mn);
      float rs = 0.f;
      #pragma unroll
      for (int jt = 0; jt < 2; ++jt) {
        const float p = __expf(st[jt][v] - mn);
        st[jt][v] = p;
        rs += p;
      }
      #pragma unroll
      for (int off = 8; off >= 1; off >>= 1)
        rs += __shfl_xor(rs, off, 16);
      mrow[v]  = mn;
      lrow[v]  = lrow[v] * a + rs;
      alpha[v] = a;
    }
    #pragma unroll
    for (int v = 0; v < 8; ++v) { acc0[v] *= alpha[v]; acc1[v] *= alpha[v]; }

    // ---- P -> LDS (wave-private rows; same-wave DS ops are in order) ----
    #pragma unroll
    for (int jt = 0; jt < 2; ++jt)
      #pragma unroll
      for (int v = 0; v < 8; ++v)
        lds_p[(istrip + v + 8 * half) * 33 + jt * 16 + l16] = st[jt][v];

    // ---- O += P * V^T: M=i, K-dim=j(32), N=d(32) ----
    #pragma unroll
    for (int kk = 0; kk < kJB; kk += 4) {
      v2f a, b0, b1;
      a.x  = lds_p[(istrip + l16) * 33 + kk + half * 2 + 0];
      a.y  = lds_p[(istrip + l16) * 33 + kk + half * 2 + 1];
      b0.x = lv[(l16)      * 34 + kk + half * 2 + 0];
      b0.y = lv[(l16)      * 34 + kk + half * 2 + 1];
      acc0 = WMMA_F32(a, b0, acc0);
      b1.x = lv[(16 + l16) * 34 + kk + half * 2 + 0];
      b1.y = lv[(16 + l16) * 34 + kk + half * 2 + 1];
      acc1 = WMMA_F32(a, b1, acc1);
    }
  }

  // ---- epilogue: stage O^T in LDS (stride 132, conflict-free), then
  //      coalesced float4 row stores: ctx[b][h*32+d][i0..i0+127] ----
  __syncthreads();                        // all waves done with lds_p strips
  #pragma unroll
  for (int v = 0; v < 8; ++v) {
    const float inv = 1.0f / lrow[v];
    const int ig = istrip + v + 8 * half;
    lds_p[(l16)      * 132 + ig] = acc0[v] * inv;
    lds_p[(16 + l16) * 132 + ig] = acc1[v] * inv;
  }
  __syncthreads();
  float* cb = ctx + ((size_t)b * kC + h * kHD) * kN + i0;
  #pragma unroll
  for (int idx = tid; idx < 1024; idx += 256) {
    const int d = idx >> 5, c4 = (idx & 31) << 2;
    *(float4*)(cb + (size_t)d * kN + c4) = *(const float4*)&lds_p[d * 132 + c4];
  }
}

// ---------------------------------------------------------------------------
extern "C" void kernel_launch(void* const* d_in, const int* in_sizes, int n_in,
                              void* d_out, int out_size, void* d_ws, size_t ws_size,
                              hipStream_t stream)
{
  const float* x      = (const float*)d_in[0];   // (4,256,48,48)
  const float* w_qkv  = (const float*)d_in[1];   // (768,256)
  const float* w_proj = (const float*)d_in[2];   // (256,256)
  const float* b_proj = (const float*)d_in[3];   // (256,)
  float* out = (float*)d_out;                    // (4,256,48,48)

  // workspace: qkv (28.3MB) + ctx (9.4MB)
  float* qkv = (float*)d_ws;
  float* ctx = qkv + (size_t)kB * 3 * kC * kN;

  dim3 blk(256);
  gemm_wmma<<<dim3(kN / 128, 768 / 64, kB), blk, 0, stream>>>(w_qkv, x, nullptr, qkv, 768);
  attn_wmma<<<dim3(kN / 128, kNH, kB), blk, 0, stream>>>(qkv, ctx);
  gemm_wmma<<<dim3(kN / 128, 256 / 64, kB), blk, 0, stream>>>(w_proj, ctx, b_proj, out, 256);
}